// HyperSAGE_19061064859834
// MI455X (gfx1250) — compile-verified
//
#include <hip/hip_runtime.h>

#define NN 50000
#define NE 1600000
#define IN_DIM 64
#define HIDDEN 128
#define OUT_DIM 64

typedef float v2f __attribute__((ext_vector_type(2)));
typedef float v8f __attribute__((ext_vector_type(8)));

// ---------------------------------------------------------------- zero
__global__ void zero_kernel(float* __restrict__ p, int n) {
    int i = blockIdx.x * blockDim.x + threadIdx.x;
    if (i < n) p[i] = 0.0f;
}

// ---------------------------------------------------------------- edge scatter
// One group of (F/4) threads per edge; each thread gathers float4 from the
// source row and atomically accumulates into the destination row (L2 atomics).
template <int F, bool COUNT>
__global__ void scatter_kernel(const float* __restrict__ feat,
                               const int* __restrict__ ei,
                               float* __restrict__ agg,
                               float* __restrict__ cnt) {
    constexpr int TPE = F / 4;
    long tid = (long)blockIdx.x * blockDim.x + threadIdx.x;
    long e = tid / TPE;
    int  f = (int)(tid % TPE) * 4;
    if (e >= NE) return;
    int s = ei[e];
    int d = ei[NE + e];
    const float4 v = *(const float4*)(feat + (long)s * F + f);
    float* ap = agg + (long)d * F + f;
    atomicAdd(ap + 0, v.x);
    atomicAdd(ap + 1, v.y);
    atomicAdd(ap + 2, v.z);
    atomicAdd(ap + 3, v.w);
    if (COUNT && (tid % TPE) == 0) atomicAdd(cnt + d, 1.0f);
}

// ---------------------------------------------------------------- fused SAGE GEMM
// out[m,n] = act( (agg[m,:]/max(cnt[m],1)) @ wl + xin[m,:] @ wr + bias )
// 256 threads = 8 waves; each wave owns one 16x16 output tile via
// V_WMMA_F32_16X16X4_F32 (K stepped by 4). Weights staged in LDS.
template <int K, int N, bool RELU>
__global__ void sage_gemm(const float* __restrict__ agg,
                          const float* __restrict__ xin,
                          const float* __restrict__ cnt,
                          const float* __restrict__ wl,
                          const float* __restrict__ wr,
                          const float* __restrict__ bias,
                          float* __restrict__ out) {
    constexpr int NT  = N / 16;   // n-tiles covered by the 8 waves
    constexpr int MPB = 8 / NT;   // m-tiles per block

    __shared__ float lwl[K * N];
    __shared__ float lwr[K * N];
    __shared__ float lb[N];

    for (int i = threadIdx.x; i < K * N; i += 256) {
        lwl[i] = wl[i];
        lwr[i] = wr[i];
    }
    for (int i = threadIdx.x; i < N; i += 256) lb[i] = bias[i];
    __syncthreads();

    const int wave = threadIdx.x >> 5;
    const int lane = threadIdx.x & 31;
    const int lp   = lane & 15;         // position within half-wave
    const int kh   = (lane >> 4) << 1;  // K sub-offset: 0 (lanes 0-15) / 2 (16-31)

    const int mtile = blockIdx.x * MPB + wave / NT;
    const int ntile = wave % NT;
    const int m0 = mtile * 16;
    const int n0 = ntile * 16;

    // WMMA needs full EXEC: clamp load row, predicate only the stores.
    int rowA = m0 + lp;
    if (rowA >= NN) rowA = NN - 1;
    const float invc = 1.0f / fmaxf(cnt[rowA], 1.0f);
    const float* aggRow = agg + (long)rowA * K;
    const float* xRow   = xin + (long)rowA * K;

    v8f acc = {};
#pragma unroll
    for (int k = 0; k < K; k += 4) {
        v2f a, b;
        // aggregated-neighbor term (mean folded in as per-row scale)
        a.x = aggRow[k + kh + 0] * invc;
        a.y = aggRow[k + kh + 1] * invc;
        b.x = lwl[(k + kh + 0) * N + n0 + lp];
        b.y = lwl[(k + kh + 1) * N + n0 + lp];
        acc = __builtin_amdgcn_wmma_f32_16x16x4_f32(false, a, false, b,
                                                    (short)0, acc, false, false);
        // self term
        a.x = xRow[k + kh + 0];
        a.y = xRow[k + kh + 1];
        b.x = lwr[(k + kh + 0) * N + n0 + lp];
        b.y = lwr[(k + kh + 1) * N + n0 + lp];
        acc = __builtin_amdgcn_wmma_f32_16x16x4_f32(false, a, false, b,
                                                    (short)0, acc, false, false);
    }

    const float bv = lb[n0 + lp];
    const int mBase = m0 + ((lane >> 4) << 3);  // C layout: VGPR v -> M = v + 8*(lane/16)
#pragma unroll
    for (int v = 0; v < 8; ++v) {
        int m = mBase + v;
        float r = acc[v] + bv;
        if (RELU) r = fmaxf(r, 0.0f);
        if (m < NN) out[(long)m * N + n0 + lp] = r;
    }
}

// ---------------------------------------------------------------- launch
extern "C" void kernel_launch(void* const* d_in, const int* in_sizes, int n_in,
                              void* d_out, int out_size, void* d_ws, size_t ws_size,
                              hipStream_t stream) {
    const float* x   = (const float*)d_in[0];
    const int*   ei  = (const int*)d_in[1];
    const float* wl1 = (const float*)d_in[2];
    const float* wr1 = (const float*)d_in[3];
    const float* b1  = (const float*)d_in[4];
    const float* wl2 = (const float*)d_in[5];
    const float* wr2 = (const float*)d_in[6];
    const float* b2  = (const float*)d_in[7];
    float* out = (float*)d_out;

    // workspace layout (floats): agg1 | agg2 | cnt | h   (~64.2 MB total)
    float* agg1 = (float*)d_ws;
    float* agg2 = agg1 + (size_t)NN * IN_DIM;
    float* cnt  = agg2 + (size_t)NN * HIDDEN;
    float* h    = cnt + NN;

    // zero agg1 + agg2 + cnt (contiguous region)
    const int zn = NN * (IN_DIM + HIDDEN + 1);
    zero_kernel<<<(zn + 255) / 256, 256, 0, stream>>>(agg1, zn);

    // ---- layer 1 ----
    {
        long tot = (long)NE * (IN_DIM / 4);
        scatter_kernel<IN_DIM, true>
            <<<(int)((tot + 255) / 256), 256, 0, stream>>>(x, ei, agg1, cnt);
    }
    sage_gemm<IN_DIM, HIDDEN, true>
        <<<NN / 16, 256, 0, stream>>>(agg1, x, cnt, wl1, wr1, b1, h);

    // ---- layer 2 ----
    {
        long tot = (long)NE * (HIDDEN / 4);
        scatter_kernel<HIDDEN, false>
            <<<(int)((tot + 255) / 256), 256, 0, stream>>>(h, ei, agg2, nullptr);
    }
    sage_gemm<HIDDEN, OUT_DIM, false>
        <<<(NN / 16 + 1) / 2, 256, 0, stream>>>(agg2, h, cnt, wl2, wr2, b2, out);
}